// CIFARDiffusionLayer_5394478923805
// MI455X (gfx1250) — compile-verified
//
#include <hip/hip_runtime.h>
#include <stdint.h>

// ============================================================================
// ADI diffusion (4 steps, 12 tridiagonal sweeps) fused into ONE pass over u.
//
// Roofline: 402 MB min traffic @ 23.3 TB/s = ~17 us. Thomas algorithm keeps
// FLOPs at ~4.3 GFLOP (trivial). Batch-independent Thomas coefficients
// (a, 1/denom, c_star) are precomputed into d_ws (288 KB, L2-resident).
//
// CDNA5 paths used:
//   - global_load_async_to_lds_b32 / global_store_async_from_lds_b32
//     (async tensor/LDS data movers, ASYNCcnt) for tile staging
//   - s_wait_asynccnt
//   - wave32 lane=batch layout; v_readlane coefficient broadcast (VALU, not DS)
// ============================================================================

#define DTK   0.15f
#define EPSK  1e-6f
#define SK    32
#define CK    3
#define BK    16384
#define IMGS  32          // images per block (== wave32 lanes)
#define PADW  1025        // LDS words per image; 1025 % 64 == 1 -> conflict-free
#define THR   256

// ws float layout: [dir(2)][arr(3: a, inv, cstar)][step(4)][c(3)][line(32)][i(32)]
//   dir offset = dir*36864, arr offset = arr*12288
__device__ __forceinline__ int coefBase(int dir, int arr, int s, int c, int line) {
    return dir * 36864 + arr * 12288 + ((s * CK + c) * SK + line) * SK;
}

// ---------------------------------------------------------------------------
// Precompute Thomas coefficients for every (dir, step, channel, line).
// dir 0 = X (solve along w, dt = DT/2, smooth alpha along w)
// dir 1 = Y (solve along h, dt = DT,   smooth beta  along h)
// Exactly mirrors the reference: denom = b - a*cp + EPS each step; cstar[31]=0.
// ---------------------------------------------------------------------------
__global__ void precompute_coeffs(const float* __restrict__ aBase,
                                  const float* __restrict__ bBase,
                                  const float* __restrict__ aTc,
                                  const float* __restrict__ bTc,
                                  float* __restrict__ ws) {
    const int blk = blockIdx.x;          // 0..23
    const int dir = blk / 12;
    const int rem = blk % 12;
    const int s   = rem / CK;
    const int c   = rem % CK;
    const int l   = threadIdx.x;         // line 0..31
    const float t  = s * DTK;
    const float dt = (dir == 0) ? (0.5f * DTK) : DTK;
    const float* base = (dir == 0) ? aBase : bBase;
    const float* tc   = (dir == 0) ? aTc   : bTc;

    float* A  = ws + coefBase(dir, 0, s, c, l);
    float* I  = ws + coefBase(dir, 1, s, c, l);
    float* Cs = ws + coefBase(dir, 2, s, c, l);

    auto val = [&](int j) -> float {   // field value at position j along the line
        int h, w;
        if (dir == 0) { h = l; w = j; } else { h = j; w = l; }
        int idx = (c * SK + h) * SK + w;
        float v = base[idx] + tc[idx] * t;
        return fmaxf(v, EPSK);
    };

    float cp = 0.f;
    for (int i = 0; i < SK; ++i) {
        int jm = (i == 0) ? 0 : i - 1;
        int jp = (i == SK - 1) ? SK - 1 : i + 1;
        float sm = (val(jm) + val(i) + val(jp)) / 3.0f;   // replicate-pad 3-tap mean
        float cf = sm * dt;                                // DX^2 == 1
        float a  = -cf;
        float b  = ((i == 0) || (i == SK - 1)) ? (1.f + cf) : (1.f + 2.f * cf);
        float denom = b - a * cp + EPSK;
        float inv = 1.0f / denom;
        cp = (-cf) * inv;                                  // c_i / denom
        A[i]  = a;
        I[i]  = inv;
        Cs[i] = (i == SK - 1) ? 0.f : cp;                  // reference zeroes last
    }
}

// broadcast lane i of v to all lanes (pure VALU/SALU, keeps DS pipe free)
__device__ __forceinline__ float bcast(float v, int i) {
    return __int_as_float(__builtin_amdgcn_readlane(__float_as_int(v), i));
}

// ---------------------------------------------------------------------------
// Main fused kernel. One block = one channel x 32 batch images, all resident
// in LDS (32 * 1025 words = 131.2 KB -> 2 workgroups per 320 KB WGP).
// lane = image (batch) index; each wave owns 4 lines per sweep.
// ---------------------------------------------------------------------------
__global__ __launch_bounds__(THR)
void diffusion_main(const float* __restrict__ u,
                    const float* __restrict__ ws,
                    const float* __restrict__ coupling,
                    float* __restrict__ out) {
    extern __shared__ float sm[];
    const int tid  = threadIdx.x;
    const int lane = tid & 31;
    const int wave = tid >> 5;
    const int c    = blockIdx.x / (BK / IMGS);
    const int bt   = blockIdx.x % (BK / IMGS);
    const long b0  = (long)bt * IMGS;

    const float* gbase = u   + ((size_t)(b0 * CK + c)) * (SK * SK);
    float*       obase = out + ((size_t)(b0 * CK + c)) * (SK * SK);
    const float dc = coupling[c * CK + c];   // einsum 'cc,bchw' = diagonal scale

    // ---- stage tile into LDS via CDNA5 async data mover (ASYNCcnt) ----
    // Each op: 32 lanes x 4B = 128B contiguous global -> conflict-free LDS.
    for (int q = wave; q < IMGS * 32; q += 8) {
        int img = q >> 5, k = q & 31;
        uint32_t voff = (uint32_t)((img * (CK * SK * SK) + k * 32 + lane) * 4);
        uint32_t loff = (uint32_t)(uintptr_t)(sm + img * PADW + k * 32 + lane);
        asm volatile("global_load_async_to_lds_b32 %0, %1, %2"
                     :: "v"(loff), "v"(voff), "s"(gbase) : "memory");
    }
    asm volatile("s_wait_asynccnt 0x0" ::: "memory");
    __syncthreads();

    // ---- one tridiagonal sweep over all 32 lines of this tile ----
    // dir 0: line = h, element i = w (stride 1 word)
    // dir 1: line = w, element i = h (stride 32 words)
    // om: output multiplier (channel coupling folded into 2nd X-sweep of a step)
    auto sweep = [&](int dir, int s, float om) {
        const float* pA = ws + coefBase(dir, 0, s, c, 0);
        const int stride = (dir == 0) ? 1 : SK;
        for (int r = 0; r < 4; ++r) {
            const int line = wave * 4 + r;
            const float* lA = pA + line * SK;
            float aV = lA[lane];            // lane i holds a[i]      (coalesced)
            float iV = lA[12288 + lane];    // lane i holds 1/denom[i]
            float cV = lA[24576 + lane];    // lane i holds cstar[i]
            const int base = lane * PADW + ((dir == 0) ? line * SK : line);

            float dp[SK];
            float prev = 0.f;
            #pragma unroll
            for (int i = 0; i < SK; ++i) {          // forward sweep
                float d = sm[base + i * stride];
                prev = (d - bcast(aV, i) * prev) * bcast(iV, i);
                dp[i] = prev;
            }
            float xn = 0.f;
            #pragma unroll
            for (int i = SK - 1; i >= 0; --i) {     // backward substitution
                xn = dp[i] - bcast(cV, i) * xn;
                sm[base + i * stride] = xn * om;
            }
        }
        __syncthreads();
    };

    // 4 steps: X(dt/2), Y(dt), X(dt/2) then diagonal channel scale (folded)
    #pragma unroll
    for (int s = 0; s < 4; ++s) {
        sweep(0, s, 1.f);
        sweep(1, s, 1.f);
        sweep(0, s, dc);
    }

    // ---- stream tile back via async LDS->global store ----
    for (int q = wave; q < IMGS * 32; q += 8) {
        int img = q >> 5, k = q & 31;
        uint32_t voff = (uint32_t)((img * (CK * SK * SK) + k * 32 + lane) * 4);
        uint32_t loff = (uint32_t)(uintptr_t)(sm + img * PADW + k * 32 + lane);
        asm volatile("global_store_async_from_lds_b32 %0, %1, %2"
                     :: "v"(voff), "v"(loff), "s"(obase) : "memory");
    }
    asm volatile("s_wait_asynccnt 0x0" ::: "memory");
}

// ---------------------------------------------------------------------------
extern "C" void kernel_launch(void* const* d_in, const int* in_sizes, int n_in,
                              void* d_out, int out_size, void* d_ws, size_t ws_size,
                              hipStream_t stream) {
    const float* u     = (const float*)d_in[0];
    const float* aBase = (const float*)d_in[1];
    const float* bBase = (const float*)d_in[2];
    const float* aTc   = (const float*)d_in[3];
    const float* bTc   = (const float*)d_in[4];
    const float* coup  = (const float*)d_in[5];
    float* ws  = (float*)d_ws;     // needs 73728 floats = 288 KB
    float* out = (float*)d_out;

    precompute_coeffs<<<24, 32, 0, stream>>>(aBase, bBase, aTc, bTc, ws);

    const size_t lds = (size_t)IMGS * PADW * sizeof(float);   // 131,200 B
    diffusion_main<<<(BK / IMGS) * CK, THR, lds, stream>>>(u, ws, coup, out);
}